// FraudGNN_2508260901300
// MI455X (gfx1250) — compile-verified
//
#include <hip/hip_runtime.h>
#include <hip/hip_bf16.h>

typedef __attribute__((ext_vector_type(2))) float v2f;
typedef __attribute__((ext_vector_type(8))) float v8f;

// ---------------------------------------------------------------------------
// Zero-fill
// ---------------------------------------------------------------------------
__global__ void zero_f32(float* __restrict__ p, int n) {
    int i = blockIdx.x * blockDim.x + threadIdx.x;
    if (i < n) p[i] = 0.0f;
}

// ---------------------------------------------------------------------------
// Edge scatter: agg[d] += x[s] (128 features), cnt[d] += 1. One wave per edge.
// ---------------------------------------------------------------------------
__global__ void scatter_accum_128(const long long* __restrict__ ei,
                                  const float* __restrict__ x,
                                  float* __restrict__ agg,
                                  float* __restrict__ cnt, int nE) {
    int gid  = blockIdx.x * blockDim.x + threadIdx.x;
    int e    = gid >> 5;
    int lane = gid & 31;
    if (e >= nE) return;
    long long s = ei[e];
    long long d = ei[(long long)nE + e];
    const float* xs = x + s * 128;
    float* ad = agg + d * 128;
#pragma unroll
    for (int j = 0; j < 4; ++j) {
        int f = lane + j * 32;
        atomicAdd(&ad[f], xs[f]);
    }
    if (lane == 0) atomicAdd(&cnt[d], 1.0f);
}

// ---------------------------------------------------------------------------
// Edge scatter: agg[d] += h[s] (64 features). One wave per edge.
// ---------------------------------------------------------------------------
__global__ void scatter_accum_64(const long long* __restrict__ ei,
                                 const float* __restrict__ h,
                                 float* __restrict__ agg, int nE) {
    int gid  = blockIdx.x * blockDim.x + threadIdx.x;
    int e    = gid >> 5;
    int lane = gid & 31;
    if (e >= nE) return;
    long long s = ei[e];
    long long d = ei[(long long)nE + e];
    const float* hs = h + s * 64;
    float* ad = agg + d * 64;
#pragma unroll
    for (int j = 0; j < 2; ++j) {
        int f = lane + j * 32;
        atomicAdd(&ad[f], hs[f]);
    }
}

// ---------------------------------------------------------------------------
// Fused SAGE GEMM via V_WMMA_F32_16X16X4_F32 (exact fp32 matrix pipe):
//   out[16x64 tile] = relu( (agg/max(cnt,1)) @ Wl + bias + xroot @ Wr )
// Weights are staged into LDS once per block; B fragments come from ds_load.
// One wave computes one 16-row tile across all 64 output columns.
// EXEC is all-1s for every WMMA (whole-wave tile guard only).
// ---------------------------------------------------------------------------
template <int KDIM>
__global__ __launch_bounds__(256) void sage_gemm_wmma(
        const float* __restrict__ agg, const float* __restrict__ cnt,
        const float* __restrict__ xroot,
        const float* __restrict__ Wl, const float* __restrict__ Wr,
        const float* __restrict__ bias,
        float* __restrict__ out, int n) {
    __shared__ float sWl[KDIM * 64];
    __shared__ float sWr[KDIM * 64];
    for (int i = threadIdx.x; i < KDIM * 64; i += 256) {
        sWl[i] = Wl[i];
        sWr[i] = Wr[i];
    }
    __syncthreads();

    int wave = (blockIdx.x * blockDim.x + threadIdx.x) >> 5;
    int lane = threadIdx.x & 31;
    int tile = wave;
    if (tile * 16 >= n) return;                 // uniform per wave (post-barrier)

    int l16  = lane & 15;
    int koff = (lane >> 4) << 1;                // 0 for lanes 0-15, 2 for 16-31
    int row  = tile * 16 + l16;
    if (row >= n) row = n - 1;                  // clamp (stores are masked)

    float invc = 1.0f / fmaxf(cnt[row], 1.0f);

    v8f acc[4];
#pragma unroll
    for (int t = 0; t < 4; ++t)
#pragma unroll
        for (int i = 0; i < 8; ++i) acc[t][i] = 0.0f;

    const float* arow = agg   + (long long)row * KDIM;
    const float* xrow = xroot + (long long)row * KDIM;

#pragma unroll 4
    for (int k = 0; k < KDIM; k += 4) {
        int kk = k + koff;
        v2f a1 = *(const v2f*)(arow + kk);      // A frag (neighbor mean), scaled
        a1.x *= invc; a1.y *= invc;
        v2f a2 = *(const v2f*)(xrow + kk);      // A frag (root path)
#pragma unroll
        for (int t = 0; t < 4; ++t) {
            int col = t * 16 + l16;
            v2f bl; bl.x = sWl[kk * 64 + col];
                    bl.y = sWl[(kk + 1) * 64 + col];
            acc[t] = __builtin_amdgcn_wmma_f32_16x16x4_f32(
                false, a1, false, bl, (short)0, acc[t], false, false);
            v2f br; br.x = sWr[kk * 64 + col];
                    br.y = sWr[(kk + 1) * 64 + col];
            acc[t] = __builtin_amdgcn_wmma_f32_16x16x4_f32(
                false, a2, false, br, (short)0, acc[t], false, false);
        }
    }

    // Epilogue: bias + relu + store. C/D layout: VGPR i -> M = i (+8 for hi half)
    int rbase = tile * 16 + ((lane >> 4) << 3);
#pragma unroll
    for (int t = 0; t < 4; ++t) {
        int col = t * 16 + l16;
        float bv = bias[col];
#pragma unroll
        for (int i = 0; i < 8; ++i) {
            int r = rbase + i;
            if (r < n) {
                float v = acc[t][i] + bv;
                out[(long long)r * 64 + col] = fmaxf(v, 0.0f);
            }
        }
    }
}

// ---------------------------------------------------------------------------
// GAT prep: g = h2 @ Wg (64x2), attention logits, init segment-max with the
// self-loop contribution. One thread per node.
// ---------------------------------------------------------------------------
__global__ void gat_prep(const float* __restrict__ h2, const float* __restrict__ Wg,
                         const float* __restrict__ att_src, const float* __restrict__ att_dst,
                         float* __restrict__ g, float* __restrict__ a_s,
                         float* __restrict__ a_d, float* __restrict__ mmax, int n) {
    int i = blockIdx.x * blockDim.x + threadIdx.x;
    if (i >= n) return;
    const float* hv = h2 + (long long)i * 64;
    float g0 = 0.0f, g1 = 0.0f;
#pragma unroll 8
    for (int k = 0; k < 64; ++k) {
        float hk = hv[k];
        g0 += hk * Wg[k * 2];
        g1 += hk * Wg[k * 2 + 1];
    }
    g[2 * i]     = g0;
    g[2 * i + 1] = g1;
    float as = g0 * att_src[0] + g1 * att_src[1];
    float ad = g0 * att_dst[0] + g1 * att_dst[1];
    a_s[i] = as;
    a_d[i] = ad;
    float eself = as + ad;
    eself = eself > 0.0f ? eself : 0.2f * eself;   // leaky_relu(0.2)
    mmax[i] = eself;                               // self-loop seeds the max
}

__device__ __forceinline__ void atomicMaxF32(float* addr, float val) {
    int* ai = (int*)addr;
    int old = __float_as_int(*addr);
    while (__int_as_float(old) < val) {
        int prev = atomicCAS(ai, old, __float_as_int(val));
        if (prev == old) break;
        old = prev;
    }
}

__global__ void gat_edge_max(const long long* __restrict__ ei,
                             const float* __restrict__ a_s, const float* __restrict__ a_d,
                             float* __restrict__ mmax, int nE) {
    int e = blockIdx.x * blockDim.x + threadIdx.x;
    if (e >= nE) return;
    long long s = ei[e];
    long long d = ei[(long long)nE + e];
    float v = a_s[s] + a_d[d];
    v = v > 0.0f ? v : 0.2f * v;
    atomicMaxF32(&mmax[d], v);
}

// Fused numerator + denominator (alpha = w/denom factors out of the sum).
__global__ void gat_edge_sum(const long long* __restrict__ ei,
                             const float* __restrict__ a_s, const float* __restrict__ a_d,
                             const float* __restrict__ mmax, const float* __restrict__ g,
                             float* __restrict__ denom, float* __restrict__ num, int nE) {
    int e = blockIdx.x * blockDim.x + threadIdx.x;
    if (e >= nE) return;
    long long s = ei[e];
    long long d = ei[(long long)nE + e];
    float v = a_s[s] + a_d[d];
    v = v > 0.0f ? v : 0.2f * v;
    float w = __expf(v - mmax[d]);
    atomicAdd(&denom[d], w);
    atomicAdd(&num[2 * d],     w * g[2 * s]);
    atomicAdd(&num[2 * d + 1], w * g[2 * s + 1]);
}

// Final: fold in self-loop, divide, bias, log_softmax over 2 classes.
__global__ void gat_final(const float* __restrict__ a_s, const float* __restrict__ a_d,
                          const float* __restrict__ mmax, const float* __restrict__ g,
                          const float* __restrict__ denom, const float* __restrict__ num,
                          const float* __restrict__ bg, float* __restrict__ out, int n) {
    int i = blockIdx.x * blockDim.x + threadIdx.x;
    if (i >= n) return;
    float v = a_s[i] + a_d[i];
    v = v > 0.0f ? v : 0.2f * v;
    float ws  = __expf(v - mmax[i]);
    float den = fmaxf(denom[i] + ws, 1e-16f);
    float n0  = num[2 * i]     + ws * g[2 * i];
    float n1  = num[2 * i + 1] + ws * g[2 * i + 1];
    float o0  = n0 / den + bg[0];
    float o1  = n1 / den + bg[1];
    float mx  = fmaxf(o0, o1);
    float lse = mx + __logf(__expf(o0 - mx) + __expf(o1 - mx));
    out[2 * i]     = o0 - lse;
    out[2 * i + 1] = o1 - lse;
}

// ---------------------------------------------------------------------------
// Host launch
// ---------------------------------------------------------------------------
extern "C" void kernel_launch(void* const* d_in, const int* in_sizes, int n_in,
                              void* d_out, int out_size, void* d_ws, size_t ws_size,
                              hipStream_t stream) {
    const float*     x       = (const float*)d_in[0];
    const long long* ei      = (const long long*)d_in[1];   // int64 [2,E]
    const float*     W1l     = (const float*)d_in[2];
    const float*     b1l     = (const float*)d_in[3];
    const float*     W1r     = (const float*)d_in[4];
    const float*     W2l     = (const float*)d_in[5];
    const float*     b2l     = (const float*)d_in[6];
    const float*     W2r     = (const float*)d_in[7];
    const float*     Wg      = (const float*)d_in[8];
    const float*     bg      = (const float*)d_in[9];
    const float*     att_src = (const float*)d_in[10];
    const float*     att_dst = (const float*)d_in[11];
    float*           out     = (float*)d_out;

    const int n = in_sizes[0] / 128;    // 50000
    const int e = in_sizes[1] / 2;      // 800000

    // Workspace carve-out (floats)
    float* ws = (float*)d_ws;
    size_t off = 0;
    float* agg1  = ws + off; off += (size_t)n * 128;
    float* cnt   = ws + off; off += (size_t)n;
    float* h1    = ws + off; off += (size_t)n * 64;
    float* agg2  = ws + off; off += (size_t)n * 64;
    float* h2    = ws + off; off += (size_t)n * 64;
    float* g     = ws + off; off += (size_t)n * 2;
    float* a_s   = ws + off; off += (size_t)n;
    float* a_d   = ws + off; off += (size_t)n;
    float* mmax  = ws + off; off += (size_t)n;
    float* denom = ws + off; off += (size_t)n;
    float* num   = ws + off; off += (size_t)n * 2;
    (void)ws_size; (void)n_in; (void)out_size;

    auto zero = [&](float* p, size_t len) {
        int blocks = (int)((len + 255) / 256);
        zero_f32<<<blocks, 256, 0, stream>>>(p, (int)len);
    };
    zero(agg1, (size_t)n * 128);
    zero(cnt, n);
    zero(agg2, (size_t)n * 64);
    zero(denom, n);
    zero(num, (size_t)n * 2);

    // Layer 1: scatter + fused WMMA GEMM (K = 128)
    {
        int blocks = (e * 32 + 255) / 256;
        scatter_accum_128<<<blocks, 256, 0, stream>>>(ei, x, agg1, cnt, e);
    }
    int tiles   = (n + 15) / 16;
    int gblocks = (tiles + 7) / 8;      // 8 waves per 256-thread block
    sage_gemm_wmma<128><<<gblocks, 256, 0, stream>>>(agg1, cnt, x, W1l, W1r, b1l, h1, n);

    // Layer 2: scatter + fused WMMA GEMM (K = 64)
    {
        int blocks = (e * 32 + 255) / 256;
        scatter_accum_64<<<blocks, 256, 0, stream>>>(ei, h1, agg2, e);
    }
    sage_gemm_wmma<64><<<gblocks, 256, 0, stream>>>(agg2, cnt, h1, W2l, W2r, b2l, h2, n);

    // GAT head
    gat_prep<<<(n + 255) / 256, 256, 0, stream>>>(h2, Wg, att_src, att_dst, g, a_s, a_d, mmax, n);
    gat_edge_max<<<(e + 255) / 256, 256, 0, stream>>>(ei, a_s, a_d, mmax, e);
    gat_edge_sum<<<(e + 255) / 256, 256, 0, stream>>>(ei, a_s, a_d, mmax, g, denom, num, e);
    gat_final<<<(n + 255) / 256, 256, 0, stream>>>(a_s, a_d, mmax, g, denom, num, bg, out, n);
}